// TPVM_block_72344429134359
// MI455X (gfx1250) — compile-verified
//
#include <hip/hip_runtime.h>
#include <hip/hip_bf16.h>
#include <math.h>

// ---------------- WMMA helpers (CDNA5 fp32 path: V_WMMA_F32_16X16X4_F32) ---
typedef float v2f __attribute__((ext_vector_type(2)));
typedef float v8f __attribute__((ext_vector_type(8)));

__device__ inline v8f v8zero() {
    v8f z;
#pragma unroll
    for (int q = 0; q < 8; ++q) z[q] = 0.0f;
    return z;
}

// D = A(16x4,f32) * B(4x16,f32) + C(16x16,f32)
// Lane mapping (ISA 7.12.2, wave32):
//   A: lanes0-15 -> M=lane, regs {K0,K1}; lanes16-31 -> M=lane-16, regs {K2,K3}
//   B: lanes0-15 -> N=lane, regs {K0,K1}; lanes16-31 -> N=lane-16, regs {K2,K3}
//   C/D: reg g -> (M=g, N=l16) for half 0, (M=g+8, N=l16) for half 1
__device__ inline v8f wmma4(v2f a, v2f b, v8f c) {
    return __builtin_amdgcn_wmma_f32_16x16x4_f32(false, a, false, b, (short)0, c,
                                                 false, false);
}

// ---------------- Problem constants ---------------------------------------
#define BB 16
#define CI 192
#define HH 64
#define WW 64
#define LL 4096            // H*W
#define CB 64              // CI/3
#define DIN 128            // D_INNER
#define DST 16             // D_STATE
#define BL (BB * LL)       // rows per branch = 65536
#define EPSV 1e-5f

// Workspace layout (floats)
#define N_XN   (BL * 64)        // 4,194,304
#define N_BIG  (BL * 128)       // 8,388,608
#define N_BC   (BL * 16)        // 1,048,576
#define N_YM   (BL * 192)       // 12,582,912
#define OFF_XN   0
#define OFF_XM   (OFF_XN + N_XN)         // also aliased as dt buffer after dwconv
#define OFF_Z    (OFF_XM + N_BIG)
#define OFF_XC   (OFF_Z  + N_BIG)
#define OFF_YS   (OFF_XC + N_BIG)
#define OFF_BM   (OFF_YS + N_BIG)
#define OFF_CM   (OFF_BM + N_BC)
#define OFF_YMB  (OFF_CM + N_BC)
#define OFF_AEXP (OFF_YMB + N_YM)        // 2048
#define OFF_WF   (OFF_AEXP + 2048)       // 160*128 = 20480
#define OFF_WPK  (OFF_WF + 20480)        // 9*96*64*2 = 110592

// ---------------- Kernel 0: precompute ------------------------------------
// Aexp[d][s] = -exp(A_log[d][s])
// Wf[160][128]: rows 0..127 = dt_proj_w @ x_proj_w[:4]   (fused dt path)
//               rows 128..159 = x_proj_w[4:36]           (B then C)
// wpk: conv weights packed [tap][ci/2][co][2] for float2 B-fragment loads
__global__ void prep_kernel(const float* __restrict__ A_log,
                            const float* __restrict__ dt_proj_w,
                            const float* __restrict__ x_proj_w,
                            const float* __restrict__ conv1_w,
                            float* __restrict__ Aexp,
                            float* __restrict__ Wf,
                            float* __restrict__ wpk) {
    int tid = blockIdx.x * blockDim.x + threadIdx.x;
    int nth = gridDim.x * blockDim.x;
    for (int i = tid; i < DIN * DST; i += nth) Aexp[i] = -expf(A_log[i]);
    for (int i = tid; i < 160 * 128; i += nth) {
        int n = i >> 7, k = i & 127;
        float v;
        if (n < 128) {
            v = 0.0f;
#pragma unroll
            for (int r = 0; r < 4; ++r)
                v = fmaf(dt_proj_w[n * 4 + r], x_proj_w[r * 128 + k], v);
        } else {
            v = x_proj_w[(n - 124) * 128 + k];  // rows 4..35
        }
        Wf[i] = v;
    }
    for (int i = tid; i < 9 * 96 * 64 * 2; i += nth) {
        int jj = i & 1;
        int co = (i >> 1) & 63;
        int cip = (i >> 7) % 96;
        int tap = i / 12288;
        wpk[i] = conv1_w[(co * CI + 2 * cip + jj) * 9 + tap];
    }
}

// ---------------- Kernel 1: 3x3 conv (implicit GEMM, WMMA) + bias + LN(64) -
// 1 wave per block; tile = 16 sequence positions x 64 output channels.
// branch: 0 = identity, 1 = flip W, 2 = flip H (flip applied at input read).
__global__ __launch_bounds__(32) void conv_ln_kernel(
    const float* __restrict__ x, const float* __restrict__ wpk,
    const float* __restrict__ bias, const float* __restrict__ lnw,
    const float* __restrict__ lnb, float* __restrict__ xn, int branch) {
    const int lane = threadIdx.x;
    const int half = lane >> 4, l16 = lane & 15;
    const int b = blockIdx.x >> 8;
    const int l0 = (blockIdx.x & 255) << 4;
    const int irow = l0 >> 6;   // all 16 positions share the image row
    const int j0 = l0 & 63;

    v8f acc[4];
#pragma unroll
    for (int t = 0; t < 4; ++t) acc[t] = v8zero();

    const float* xb = x + b * CI * LL;
#pragma unroll
    for (int tap = 0; tap < 9; ++tap) {
        const int dh = tap / 3 - 1, dw = tap % 3 - 1;
        const int ih = irow + dh;
        if (ih < 0 || ih >= HH) continue;        // uniform for the whole tile
        const int jw = j0 + l16 + dw;
        const bool jv = (jw >= 0) && (jw < WW);
        const int sh = (branch == 2) ? (HH - 1 - ih) : ih;
        const int sw = (branch == 1) ? (WW - 1 - ((jw < 0) ? 0 : ((jw > 63) ? 63 : jw)))
                                     : ((jw < 0) ? 0 : ((jw > 63) ? 63 : jw));
        const int spoff = sh * WW + sw;
        const float* wt = wpk + tap * 96 * 64 * 2;
        for (int kc = 0; kc < CI; kc += 4) {
            const int cip = (kc >> 1) + half;    // ci = 2*cip + jj = kc + 2*half + jj
            v2f a;
            if (jv) {
                a[0] = xb[(2 * cip + 0) * LL + spoff];
                a[1] = xb[(2 * cip + 1) * LL + spoff];
            } else {
                a[0] = 0.0f; a[1] = 0.0f;
            }
#pragma unroll
            for (int nt = 0; nt < 4; ++nt) {
                const int co = nt * 16 + l16;
                v2f bf = *reinterpret_cast<const v2f*>(wt + (cip * 64 + co) * 2);
                acc[nt] = wmma4(a, bf, acc[nt]);
            }
        }
    }

    float b4[4], w4[4], o4[4];
#pragma unroll
    for (int nt = 0; nt < 4; ++nt) {
        const int co = nt * 16 + l16;
        b4[nt] = bias[co]; w4[nt] = lnw[co]; o4[nt] = lnb[co];
    }
#pragma unroll
    for (int g = 0; g < 8; ++g) {
        float vb[4], s0 = 0.0f, s1 = 0.0f;
#pragma unroll
        for (int nt = 0; nt < 4; ++nt) {
            vb[nt] = acc[nt][g] + b4[nt];
            s0 += vb[nt];
            s1 = fmaf(vb[nt], vb[nt], s1);
        }
        // reduce across the 16 lanes of this half (rows stay separated)
        s0 += __shfl_xor(s0, 1, 32); s1 += __shfl_xor(s1, 1, 32);
        s0 += __shfl_xor(s0, 2, 32); s1 += __shfl_xor(s1, 2, 32);
        s0 += __shfl_xor(s0, 4, 32); s1 += __shfl_xor(s1, 4, 32);
        s0 += __shfl_xor(s0, 8, 32); s1 += __shfl_xor(s1, 8, 32);
        const float mean = s0 * (1.0f / 64.0f);
        const float var = s1 * (1.0f / 64.0f) - mean * mean;
        const float rstd = rsqrtf(var + EPSV);
        const int m = g + 8 * half;
        const int row = (b << 12) + l0 + m;
#pragma unroll
        for (int nt = 0; nt < 4; ++nt) {
            const int co = nt * 16 + l16;
            xn[row * 64 + co] = fmaf((vb[nt] - mean) * rstd, w4[nt], o4[nt]);
        }
    }
}

// ---------------- Kernel 2: in_proj GEMM (64 -> 256), split xm | z ---------
__global__ __launch_bounds__(128) void inproj_kernel(
    const float* __restrict__ xn, const float* __restrict__ wip,
    float* __restrict__ xm, float* __restrict__ z) {
    const int tid = threadIdx.x, wave = tid >> 5, lane = tid & 31;
    const int half = lane >> 4, l16 = lane & 15;
    const int row0 = blockIdx.x << 4;
    v8f acc[4];
#pragma unroll
    for (int t = 0; t < 4; ++t) acc[t] = v8zero();
    for (int kc = 0; kc < 64; kc += 4) {
        v2f a = *reinterpret_cast<const v2f*>(xn + (row0 + l16) * 64 + kc + 2 * half);
#pragma unroll
        for (int nt = 0; nt < 4; ++nt) {
            const int n = wave * 64 + nt * 16 + l16;
            v2f bf = *reinterpret_cast<const v2f*>(wip + n * 64 + kc + 2 * half);
            acc[nt] = wmma4(a, bf, acc[nt]);
        }
    }
#pragma unroll
    for (int nt = 0; nt < 4; ++nt) {
        const int n = wave * 64 + nt * 16 + l16;
#pragma unroll
        for (int g = 0; g < 8; ++g) {
            const int row = row0 + g + 8 * half;
            const float v = acc[nt][g];
            if (n < 128) xm[row * 128 + n] = v;
            else         z[row * 128 + n - 128] = v;
        }
    }
}

// ---------------- Kernel 3: depthwise causal conv(4) + SiLU ----------------
__global__ void dwconv_kernel(const float* __restrict__ xm,
                              const float* __restrict__ cw,
                              const float* __restrict__ cb,
                              float* __restrict__ xc, int total) {
    const int idx = blockIdx.x * blockDim.x + threadIdx.x;
    if (idx >= total) return;
    const int d = idx & 127;
    const int row = idx >> 7;
    const int l = row & 4095;
    float acc = cb[d];
#pragma unroll
    for (int k = 0; k < 4; ++k) {
        const int lp = l - 3 + k;
        if (lp >= 0) acc = fmaf(xm[(row - l + lp) * 128 + d], cw[d * 4 + k], acc);
    }
    xc[idx] = acc / (1.0f + expf(-acc));
}

// ---------------- Kernel 4: fused x_proj+dt_proj GEMM (128 -> 160) ---------
// out cols: [0,128) -> softplus(. + dt_proj_b) = dt ; [128,144) -> B ; [144,160) -> C
__global__ __launch_bounds__(64) void xproj_kernel(
    const float* __restrict__ xc, const float* __restrict__ Wf,
    const float* __restrict__ dtb, float* __restrict__ dtf,
    float* __restrict__ Bm, float* __restrict__ Cm) {
    const int tid = threadIdx.x, wave = tid >> 5, lane = tid & 31;
    const int half = lane >> 4, l16 = lane & 15;
    const int row0 = blockIdx.x << 4;
    v8f acc[5];
#pragma unroll
    for (int t = 0; t < 5; ++t) acc[t] = v8zero();
    for (int kc = 0; kc < 128; kc += 4) {
        v2f a = *reinterpret_cast<const v2f*>(xc + (row0 + l16) * 128 + kc + 2 * half);
#pragma unroll
        for (int t = 0; t < 5; ++t) {
            const int n = wave * 80 + t * 16 + l16;
            v2f bf = *reinterpret_cast<const v2f*>(Wf + n * 128 + kc + 2 * half);
            acc[t] = wmma4(a, bf, acc[t]);
        }
    }
#pragma unroll
    for (int t = 0; t < 5; ++t) {
        const int n = wave * 80 + t * 16 + l16;
#pragma unroll
        for (int g = 0; g < 8; ++g) {
            const int row = row0 + g + 8 * half;
            const float v = acc[t][g];
            if (n < 128) {
                const float u = v + dtb[n];
                dtf[row * 128 + n] = (u > 20.0f) ? u : log1pf(expf(u));
            } else if (n < 144) {
                Bm[(row << 4) + (n - 128)] = v;
            } else {
                Cm[(row << 4) + (n - 144)] = v;
            }
        }
    }
}

// ---------------- Kernel 5: selective scan ---------------------------------
// grid: 16 batches x 4 d-groups; block 512 = 32 channels x 16 states.
// Lane (d, s) carries one state; y = sum over s via shfl_xor tree.
__global__ __launch_bounds__(512) void scan_kernel(
    const float* __restrict__ dtf, const float* __restrict__ xc,
    const float* __restrict__ Bm, const float* __restrict__ Cm,
    const float* __restrict__ Aexp, float* __restrict__ ys) {
    const int tid = threadIdx.x;
    const int s = tid & 15;
    const int b = blockIdx.x >> 2;
    const int d = ((blockIdx.x & 3) << 5) + (tid >> 4);
    const float A = Aexp[(d << 4) + s];   // already -exp(A_log)
    float h = 0.0f;
    const int base = b << 12;
    for (int t = 0; t < 4096; ++t) {
        const int row = base + t;
        if ((t & 7) == 0 && t + 16 < 4096) {  // stay ahead of the dependence chain
            __builtin_prefetch(dtf + (row + 16) * 128 + d, 0, 1);
            __builtin_prefetch(xc + (row + 16) * 128 + d, 0, 1);
            __builtin_prefetch(Bm + ((row + 16) << 4) + s, 0, 1);
            __builtin_prefetch(Cm + ((row + 16) << 4) + s, 0, 1);
        }
        const float dt = dtf[row * 128 + d];
        const float xv = xc[row * 128 + d];
        const float Bt = Bm[(row << 4) + s];
        const float Ct = Cm[(row << 4) + s];
        const float dA = expf(dt * A);
        h = fmaf(dA, h, dt * xv * Bt);
        float p = h * Ct;
        p += __shfl_xor(p, 1, 32);
        p += __shfl_xor(p, 2, 32);
        p += __shfl_xor(p, 4, 32);
        p += __shfl_xor(p, 8, 32);
        if (s == 0) ys[row * 128 + d] = p;
    }
}

// ---------------- Kernel 6: gate + out_proj GEMM (128->64) + skip ----------
__global__ __launch_bounds__(32) void gateout_kernel(
    const float* __restrict__ ys, const float* __restrict__ xc,
    const float* __restrict__ z, const float* __restrict__ Dp,
    const float* __restrict__ wo, const float* __restrict__ xn,
    const float* __restrict__ skip, float* __restrict__ ym, int branch) {
    const int lane = threadIdx.x;
    const int half = lane >> 4, l16 = lane & 15;
    const int row0 = blockIdx.x << 4;
    v8f acc[4];
#pragma unroll
    for (int t = 0; t < 4; ++t) acc[t] = v8zero();
    for (int kc = 0; kc < 128; kc += 4) {
        v2f a;
#pragma unroll
        for (int jj = 0; jj < 2; ++jj) {
            const int k = kc + 2 * half + jj;
            const int rix = (row0 + l16) * 128 + k;
            const float yv = fmaf(xc[rix], Dp[k], ys[rix]);
            const float zv = z[rix];
            a[jj] = yv * (zv / (1.0f + expf(-zv)));   // y * silu(z)
        }
#pragma unroll
        for (int nt = 0; nt < 4; ++nt) {
            const int co = nt * 16 + l16;
            v2f bf = *reinterpret_cast<const v2f*>(wo + co * 128 + kc + 2 * half);
            acc[nt] = wmma4(a, bf, acc[nt]);
        }
    }
    const float sk = skip[0];
#pragma unroll
    for (int nt = 0; nt < 4; ++nt) {
        const int co = nt * 16 + l16;
#pragma unroll
        for (int g = 0; g < 8; ++g) {
            const int row = row0 + g + 8 * half;
            ym[row * 192 + branch * 64 + co] =
                fmaf(sk, xn[(row << 6) + co], acc[nt][g]);
        }
    }
}

// ---------------- Kernel 7: LN(192) + final proj GEMM (192->192) -----------
// Writes transposed directly into (B, CO, H, W).
__global__ __launch_bounds__(128) void finalproj_kernel(
    const float* __restrict__ ym, const float* __restrict__ nw,
    const float* __restrict__ nb, const float* __restrict__ pw,
    const float* __restrict__ pb, float* __restrict__ out) {
    __shared__ float rs0[16][8], rs1[16][8], smean[16], srstd[16];
    const int tid = threadIdx.x;
    const int row0 = blockIdx.x << 4;
    {   // per-row LN stats (8 threads per row)
        const int r = tid >> 3, c8 = tid & 7;
        float s0 = 0.0f, s1 = 0.0f;
        const float* p = ym + (row0 + r) * 192;
        for (int c = c8; c < 192; c += 8) {
            const float v = p[c];
            s0 += v; s1 = fmaf(v, v, s1);
        }
        rs0[r][c8] = s0; rs1[r][c8] = s1;
    }
    __syncthreads();
    if (tid < 16) {
        float s0 = 0.0f, s1 = 0.0f;
#pragma unroll
        for (int q = 0; q < 8; ++q) { s0 += rs0[tid][q]; s1 += rs1[tid][q]; }
        const float mean = s0 * (1.0f / 192.0f);
        const float var = s1 * (1.0f / 192.0f) - mean * mean;
        smean[tid] = mean;
        srstd[tid] = rsqrtf(var + EPSV);
    }
    __syncthreads();

    const int wave = tid >> 5, lane = tid & 31;
    const int half = lane >> 4, l16 = lane & 15;
    v8f acc[3];
#pragma unroll
    for (int t = 0; t < 3; ++t) acc[t] = v8zero();
    const float mu = smean[l16], rs = srstd[l16];
    for (int kc = 0; kc < 192; kc += 4) {
        v2f a;
#pragma unroll
        for (int jj = 0; jj < 2; ++jj) {
            const int k = kc + 2 * half + jj;
            const float v = ym[(row0 + l16) * 192 + k];
            a[jj] = fmaf((v - mu) * rs, nw[k], nb[k]);
        }
#pragma unroll
        for (int t = 0; t < 3; ++t) {
            const int n = wave * 48 + t * 16 + l16;
            v2f bf = *reinterpret_cast<const v2f*>(pw + n * 192 + kc + 2 * half);
            acc[t] = wmma4(a, bf, acc[t]);
        }
    }
#pragma unroll
    for (int t = 0; t < 3; ++t) {
        const int co = wave * 48 + t * 16 + l16;
#pragma unroll
        for (int g = 0; g < 8; ++g) {
            const int row = row0 + g + 8 * half;
            const int bb = row >> 12, l = row & 4095;
            out[((bb * 192 + co) << 12) + l] = acc[t][g] + pb[co];
        }
    }
}

// ---------------- Host launcher --------------------------------------------
extern "C" void kernel_launch(void* const* d_in, const int* in_sizes, int n_in,
                              void* d_out, int out_size, void* d_ws, size_t ws_size,
                              hipStream_t stream) {
    const float* x        = (const float*)d_in[0];
    const float* conv1_w  = (const float*)d_in[1];
    const float* conv1_b  = (const float*)d_in[2];
    const float* norm1_w  = (const float*)d_in[3];
    const float* norm1_b  = (const float*)d_in[4];
    const float* norm_w   = (const float*)d_in[5];
    const float* norm_b   = (const float*)d_in[6];
    const float* proj_w   = (const float*)d_in[7];
    const float* proj_b   = (const float*)d_in[8];
    const float* skip_sc  = (const float*)d_in[9];
    const float* in_prj_w = (const float*)d_in[10];
    const float* convd_w  = (const float*)d_in[11];
    const float* convd_b  = (const float*)d_in[12];
    const float* x_proj_w = (const float*)d_in[13];
    const float* dt_prj_w = (const float*)d_in[14];
    const float* dt_prj_b = (const float*)d_in[15];
    const float* A_log    = (const float*)d_in[16];
    const float* D_param  = (const float*)d_in[17];
    const float* out_pw   = (const float*)d_in[18];

    float* ws   = (float*)d_ws;
    float* xn   = ws + OFF_XN;
    float* xm   = ws + OFF_XM;     // aliased as dt after dwconv
    float* zb   = ws + OFF_Z;
    float* xcb  = ws + OFF_XC;
    float* ysb  = ws + OFF_YS;
    float* Bmb  = ws + OFF_BM;
    float* Cmb  = ws + OFF_CM;
    float* ymb  = ws + OFF_YMB;
    float* Aexp = ws + OFF_AEXP;
    float* Wf   = ws + OFF_WF;
    float* wpk  = ws + OFF_WPK;

    prep_kernel<<<128, 256, 0, stream>>>(A_log, dt_prj_w, x_proj_w, conv1_w,
                                         Aexp, Wf, wpk);

    const int mtiles = BL / 16;  // 4096
    for (int br = 0; br < 3; ++br) {
        conv_ln_kernel<<<mtiles, 32, 0, stream>>>(x, wpk, conv1_b, norm1_w,
                                                  norm1_b, xn, br);
        inproj_kernel<<<mtiles, 128, 0, stream>>>(xn, in_prj_w, xm, zb);
        dwconv_kernel<<<(BL * 128) / 256, 256, 0, stream>>>(xm, convd_w, convd_b,
                                                            xcb, BL * 128);
        xproj_kernel<<<mtiles, 64, 0, stream>>>(xcb, Wf, dt_prj_b, xm, Bmb, Cmb);
        scan_kernel<<<64, 512, 0, stream>>>(xm, xcb, Bmb, Cmb, Aexp, ysb);
        gateout_kernel<<<mtiles, 32, 0, stream>>>(ysb, xcb, zb, D_param, out_pw,
                                                  xn, skip_sc, ymb, br);
    }
    finalproj_kernel<<<mtiles, 128, 0, stream>>>(ymb, norm_w, norm_b, proj_w,
                                                 proj_b, (float*)d_out);
}